// OutputScalarNetwork_25211458027894
// MI455X (gfx1250) — compile-verified
//
#include <hip/hip_runtime.h>
#include <math.h>

#define NB 16
#define HD 64
#define CD 128
#define NN 128
#define BB 32
#define LOG2F_ 0.6931471805599453f

typedef float v2f __attribute__((ext_vector_type(2)));
typedef float v8f __attribute__((ext_vector_type(8)));

// ---------------- kernel 0: zero the per-batch scalar accumulator ----------
__global__ void osn_zero_s(float* __restrict__ s) {
    if (threadIdx.x < BB) s[threadIdx.x] = 0.0f;
}

// ---------------- kernel 1: v[b,j,h] = invsqrt(n_atoms)*sum_c W2[h,c]*rep[b,j,c]
//                  s[b] += invsqrt(n_atoms)*sum_c b2[c]*rep[b,j,c]
// one block per (b,j), 64 threads (one per h)
__global__ void __launch_bounds__(64)
osn_precompute_v(const float* __restrict__ rep,
                 const float* __restrict__ mask,
                 const float* __restrict__ W2,
                 const float* __restrict__ b2,
                 float* __restrict__ v,
                 float* __restrict__ s) {
    int bj = blockIdx.x;
    int b  = bj >> 7;      // /128
    int j  = bj & 127;
    __shared__ float repl[CD];
    __shared__ float invs_sh;
    int t = threadIdx.x;
    repl[t]      = rep[(b * NN + j) * CD + t];
    repl[t + 64] = rep[(b * NN + j) * CD + t + 64];
    if (t == 0) {
        float na = 0.0f;
        for (int k = 0; k < NN; ++k) na += mask[b * NN + k];
        invs_sh = rsqrtf(na);
    }
    __syncthreads();
    float invs = invs_sh;
    float acc = 0.0f;
    #pragma unroll 4
    for (int c = 0; c < CD; ++c) acc = fmaf(repl[c], W2[t * CD + c], acc);
    v[(b * NN + j) * HD + t] = acc * invs;
    if (t == 0) {
        float sb = 0.0f;
        for (int c = 0; c < CD; ++c) sb = fmaf(b2[c], repl[c], sb);
        atomicAdd(&s[b], sb * invs);
    }
}

// cosine^2 bump basis: centers k*0.4, step 0.4  ->  diff = 2.5*d - k
__device__ __forceinline__ float osn_cbasis(float d, float k) {
    float diff = fmaf(d, 2.5f, -k);
    float ad = fabsf(diff);
    float c = __cosf(1.5707963267948966f * diff);
    return (ad < 1.0f) ? c * c : 0.0f;
}

// ---------------- kernel 2: main fused radial-conv + activation ------------
// grid: B*(N/16) blocks, 128 threads (4 waves). Wave w owns j in [32w,32w+32).
// Per j: WMMA f32 16x16x4 chain computes h=basis@W1 for 16 rows i, then
// fused bias+relu+dot with v[b,j,:].
__global__ void __launch_bounds__(128)
osn_radial_conv(const float* __restrict__ geom,
                const float* __restrict__ mask,
                const float* __restrict__ W1,
                const float* __restrict__ b1,
                const float* __restrict__ vglob,
                const float* __restrict__ sglob,
                float* __restrict__ out) {
    __shared__ float v_lds[NN * HD];   // 32 KB
    __shared__ float g_lds[NN * 3];    // 1.5 KB
    __shared__ float part[4][32][8];   // 4 KB

    int blk  = blockIdx.x;
    int b    = blk >> 3;
    int i0   = (blk & 7) << 4;
    int tid  = threadIdx.x;
    int wave = tid >> 5;
    int lane = tid & 31;
    int ln16 = lane & 15;
    int kb   = (lane >> 4) << 1;   // K sub-offset: 0 for lanes 0-15, 2 for 16-31

    for (int idx = tid; idx < NN * HD; idx += 128)
        v_lds[idx] = vglob[b * NN * HD + idx];
    for (int idx = tid; idx < NN * 3; idx += 128)
        g_lds[idx] = geom[b * NN * 3 + idx];
    __syncthreads();

    // B operand (W1, 16x64) preloaded per-lane in 4x16 WMMA layout:
    // lane (half,ln16): rows K = 4*kk + kb (+1), col = nt*16 + ln16
    v2f w1b[4][4];
    #pragma unroll
    for (int kk = 0; kk < 4; ++kk) {
        int kr = 4 * kk + kb;
        #pragma unroll
        for (int nt = 0; nt < 4; ++nt) {
            w1b[kk][nt].x = W1[kr * HD + nt * 16 + ln16];
            w1b[kk][nt].y = W1[(kr + 1) * HD + nt * 16 + ln16];
        }
    }
    float b1v[4];
    #pragma unroll
    for (int nt = 0; nt < 4; ++nt) b1v[nt] = b1[nt * 16 + ln16];

    int irow = i0 + ln16;
    float gx = g_lds[irow * 3 + 0];
    float gy = g_lds[irow * 3 + 1];
    float gz = g_lds[irow * 3 + 2];

    float acc8[8];
    #pragma unroll
    for (int r = 0; r < 8; ++r) acc8[r] = 0.0f;

    int j0 = wave * 32;
    for (int jj = 0; jj < 32; ++jj) {
        int j = j0 + jj;
        float dx = gx - g_lds[j * 3 + 0];
        float dy = gy - g_lds[j * 3 + 1];
        float dz = gz - g_lds[j * 3 + 2];
        float d  = sqrtf(fmaf(dx, dx, fmaf(dy, dy, fmaf(dz, dz, 1e-12f))));

        // A operand: 16x4 f32 tiles of the basis (row = i-local = ln16)
        v2f a[4];
        #pragma unroll
        for (int kk = 0; kk < 4; ++kk) {
            float k0f = (float)(4 * kk + kb);
            a[kk].x = osn_cbasis(d, k0f);
            a[kk].y = osn_cbasis(d, k0f + 1.0f);
        }

        #pragma unroll
        for (int nt = 0; nt < 4; ++nt) {
            v8f c = {};
            c = __builtin_amdgcn_wmma_f32_16x16x4_f32(false, a[0], false, w1b[0][nt], (short)0, c, false, false);
            c = __builtin_amdgcn_wmma_f32_16x16x4_f32(false, a[1], false, w1b[1][nt], (short)0, c, false, false);
            c = __builtin_amdgcn_wmma_f32_16x16x4_f32(false, a[2], false, w1b[2][nt], (short)0, c, false, false);
            c = __builtin_amdgcn_wmma_f32_16x16x4_f32(false, a[3], false, w1b[3][nt], (short)0, c, false, false);
            float vj = v_lds[j * HD + nt * 16 + ln16];
            float bb = b1v[nt];
            #pragma unroll
            for (int r = 0; r < 8; ++r)
                acc8[r] = fmaf(fmaxf(c[r] + bb, 0.0f), vj, acc8[r]);
        }
    }

    #pragma unroll
    for (int r = 0; r < 8; ++r) part[wave][lane][r] = acc8[r];
    __syncthreads();

    if (tid < 16) {                 // i_local = tid
        int hi = tid >> 3;          // which lane half held this row
        int r  = tid & 7;           // accumulator slot
        float tot = 0.0f;
        #pragma unroll
        for (int w = 0; w < 4; ++w)
            #pragma unroll
            for (int g = 0; g < 16; ++g)
                tot += part[w][hi * 16 + g][r];
        float x  = tot + sglob[b];
        float sp = fmaxf(x, 0.0f) + log1pf(__expf(-fabsf(x))) - LOG2F_;
        out[b * NN + i0 + tid] = sp * mask[b * NN + i0 + tid];
    }
}

extern "C" void kernel_launch(void* const* d_in, const int* in_sizes, int n_in,
                              void* d_out, int out_size, void* d_ws, size_t ws_size,
                              hipStream_t stream) {
    const float* rep  = (const float*)d_in[0];   // [B,N,C]
    const float* geom = (const float*)d_in[1];   // [B,N,3]
    const float* mask = (const float*)d_in[2];   // [B,N]
    const float* W1   = (const float*)d_in[3];   // [NB,H]
    const float* b1   = (const float*)d_in[4];   // [H]
    const float* W2   = (const float*)d_in[5];   // [H,C]
    const float* b2   = (const float*)d_in[6];   // [C]
    float* out = (float*)d_out;                  // [B,N,1]

    float* v = (float*)d_ws;                     // B*N*H floats = 1 MB
    float* s = v + (size_t)BB * NN * HD;         // B floats

    osn_zero_s<<<1, 32, 0, stream>>>(s);
    osn_precompute_v<<<BB * NN, 64, 0, stream>>>(rep, mask, W2, b2, v, s);
    osn_radial_conv<<<BB * (NN / 16), 128, 0, stream>>>(geom, mask, W1, b1, v, s, out);
}